// HeterogenousOntologyEmbedding_36206574305717
// MI455X (gfx1250) — compile-verified
//
#include <hip/hip_runtime.h>

#define D 128
#define ASTRIDE 132  // 128 + 4 dword pad per row -> conflict-free LDS A reads

typedef float v2f __attribute__((ext_vector_type(2)));
typedef float v8f __attribute__((ext_vector_type(8)));
typedef unsigned int u32x4 __attribute__((ext_vector_type(4)));
typedef int i32x4 __attribute__((ext_vector_type(4)));
typedef int i32x8 __attribute__((ext_vector_type(8)));

// ---------------------------------------------------------------------------
// TDM: 2D tile load Global -> LDS (D# built per CDNA5 ISA 8.3/8.4).
// dim0 = row length (elems), dim1 = rows available in tensor (OOB rows -> 0),
// stride0 = row stride (elems), tile 128 x 128, optional LDS row padding.
// ---------------------------------------------------------------------------
__device__ __forceinline__ void tdm_load_2d_to_lds(
    unsigned lds_off, const void* gptr, unsigned dim0, unsigned dim1,
    unsigned stride0, unsigned tile0, unsigned tile1,
    bool pad, unsigned pad_interval_code, unsigned pad_amount_code) {
#if __has_builtin(__builtin_amdgcn_tensor_load_to_lds)
    unsigned long long ga = (unsigned long long)(uintptr_t)gptr;
    u32x4 g0;
    g0.x = 1u;                                    // count=1 (valid user D#)
    g0.y = lds_off;                               // lds_addr (bytes)
    g0.z = (unsigned)ga;                          // global_addr[31:0]
    g0.w = (unsigned)((ga >> 32) & 0x01ffffffu)   // global_addr[56:32]
           | (2u << 30);                          // type=2 ("image")
    i32x8 g1;
    unsigned d0 = (2u << 16);                     // data_size=2 -> 4 bytes
    if (pad) d0 |= (1u << 20) | (pad_interval_code << 22) | (pad_amount_code << 25);
    g1[0] = (int)d0;                              // wg_mask=0 (not in cluster)
    g1[1] = (int)((dim0 & 0xffffu) << 16);        // tensor_dim0[15:0]
    g1[2] = (int)(((dim0 >> 16) & 0xffffu) | ((dim1 & 0xffffu) << 16));
    g1[3] = (int)(((dim1 >> 16) & 0xffffu) | ((tile0 & 0xffffu) << 16));
    g1[4] = (int)(tile1 & 0xffffu);               // tile_dim1; tile_dim2=0
    g1[5] = (int)stride0;                         // tensor_dim0_stride[31:0]
    g1[6] = 0;                                    // stride0[47:32]=0; stride1=0
    g1[7] = 0;
    i32x4 z4 = {};
#if defined(__clang_major__) && (__clang_major__ >= 23)
    i32x8 z8 = {};
    __builtin_amdgcn_tensor_load_to_lds(g0, g1, z4, z4, z8, 0);
#else
    __builtin_amdgcn_tensor_load_to_lds(g0, g1, z4, z4, 0);
#endif
#else
    (void)lds_off; (void)gptr; (void)dim0; (void)dim1; (void)stride0;
    (void)tile0; (void)tile1; (void)pad; (void)pad_interval_code; (void)pad_amount_code;
#endif
}

// ---------------------------------------------------------------------------
// Utility kernels
// ---------------------------------------------------------------------------
__global__ void fill_kernel(float* __restrict__ p, long n, float v) {
    long i = (long)blockIdx.x * blockDim.x + threadIdx.x;
    long stride = (long)gridDim.x * blockDim.x;
    for (; i < n; i += stride) p[i] = v;
}

__global__ void degree_kernel(const int* __restrict__ col, long E,
                              float* __restrict__ deg) {
    long i = (long)blockIdx.x * blockDim.x + threadIdx.x;
    long stride = (long)gridDim.x * blockDim.x;
    for (; i < E; i += stride) atomicAdd(&deg[col[i]], 1.0f);
}

__global__ void rsqrt_kernel(float* __restrict__ d, long n) {
    long i = (long)blockIdx.x * blockDim.x + threadIdx.x;
    long stride = (long)gridDim.x * blockDim.x;
    for (; i < n; i += stride) d[i] = rsqrtf(d[i]);  // deg >= 1 (self loop)
}

// ---------------------------------------------------------------------------
// GEMM: h[N,128] = x[N,128] @ W[128,128] with V_WMMA_F32_16X16X4_F32.
// - A tile (128x128) staged Global->LDS by the Tensor Data Mover; TDM OOB
//   zero-fill handles the row tail; pad_enable gives a 132-dword row stride
//   (conflict-free ds_load_b64 A fragments).
// - W pre-packed in LDS in fragment order: one v2f per (kstep, ntile, lane),
//   so each B fragment is a single ds_load_b64 into an aligned VGPR pair.
// 256 threads = 8 wave32; each wave computes a 16-row x 128-col strip.
// ---------------------------------------------------------------------------
__global__ __launch_bounds__(256)
void gemm_wmma_f32_kernel(const float* __restrict__ x, const float* __restrict__ W,
                          float* __restrict__ h, int N) {
    __shared__ v2f   Wpack[32 * 8 * 32];   // 64 KB: [(kstep*8 + t)*32 + lane]
    __shared__ float Araw[128 * ASTRIDE];  // 66 KB: padded 128x128 A tile

    const int rowBase0 = blockIdx.x * 128;

#if __has_builtin(__builtin_amdgcn_tensor_load_to_lds)
    if (threadIdx.x < 32) {  // wave 0 issues the DMA (EXEC ignored by TDM)
        unsigned lds_off = (unsigned)(uintptr_t)&Araw[0];  // low 32 bits = LDS offset
        const float* gsrc = x + (long)rowBase0 * D;
        unsigned rowsAvail = (unsigned)(N - rowBase0);     // >= 1 by grid sizing
        // pad_interval code 6 -> every 128 dwords; pad_amount code 3 -> 4 dwords
        tdm_load_2d_to_lds(lds_off, gsrc, D, rowsAvail, D, 128, 128, true, 6u, 3u);
#if __has_builtin(__builtin_amdgcn_s_wait_tensorcnt)
        __builtin_amdgcn_s_wait_tensorcnt(0);
#endif
    }
#else
    // Fallback: cooperative staging with zero-fill for the row tail.
    for (int i = threadIdx.x; i < 128 * (D / 4); i += blockDim.x) {
        int r = i / (D / 4), cq = i % (D / 4);
        int gr = rowBase0 + r;
        float4 v = make_float4(0.f, 0.f, 0.f, 0.f);
        if (gr < N) v = reinterpret_cast<const float4*>(x)[(long)gr * (D / 4) + cq];
        reinterpret_cast<float4*>(&Araw[r * ASTRIDE])[cq] = v;
    }
#endif

    // Pack W (global, L2-resident) into fragment order.
    for (int p = threadIdx.x; p < 32 * 8 * 32; p += blockDim.x) {
        int lane2 = p & 31;
        int t     = (p >> 5) & 7;
        int kstep = p >> 8;
        int koff2 = (lane2 >> 4) * 2;            // lanes 0-15 -> K{0,1}; 16-31 -> K{2,3}
        int k     = kstep * 4 + koff2;
        int n     = t * 16 + (lane2 & 15);
        v2f w;
        w.x = W[k * D + n];
        w.y = W[(k + 1) * D + n];
        Wpack[p] = w;
    }
    __syncthreads();

    const int lane = threadIdx.x & 31;
    const int wave = threadIdx.x >> 5;
    const int l15  = lane & 15;
    const int koff = (lane >> 4) * 2;
    const float* arow = &Araw[(wave * 16 + l15) * ASTRIDE];

    v8f acc[8] = {};  // 16x128 output strip

    for (int kstep = 0; kstep < 32; ++kstep) {
        v2f a;  // single ds_load_b64 (conflict-free thanks to 132-dword stride)
        a.x = arow[kstep * 4 + koff];
        a.y = arow[kstep * 4 + koff + 1];
#pragma unroll
        for (int t = 0; t < 8; ++t) {
            v2f b = Wpack[(kstep * 8 + t) * 32 + lane];  // ds_load_b64, no shuffles
            acc[t] = __builtin_amdgcn_wmma_f32_16x16x4_f32(
                false, a, false, b, (short)0, acc[t], false, false);
        }
    }

    // C/D layout: VGPR v, lane l -> M = v + 8*(l>=16), N = l & 15
    const int rOff = (lane >> 4) * 8;
#pragma unroll
    for (int t = 0; t < 8; ++t) {
#pragma unroll
        for (int v = 0; v < 8; ++v) {
            int r = rowBase0 + wave * 16 + rOff + v;
            if (r < N) h[(long)r * D + t * 16 + l15] = acc[t][v];
        }
    }
}

// ---------------------------------------------------------------------------
// Edge scatter: out[col] += dinv[row]*dinv[col] * h[row].
// One wave per edge; lane handles 4 features. Next edge's h row is
// prefetched (global_prefetch_b8) since the gather is random-access.
// ---------------------------------------------------------------------------
__global__ __launch_bounds__(256)
void scatter_kernel(const float* __restrict__ h, const int* __restrict__ rows,
                    const int* __restrict__ cols, const float* __restrict__ dinv,
                    float* __restrict__ out, long E) {
    const int lane = threadIdx.x & 31;
    long wid = ((long)blockIdx.x * blockDim.x + threadIdx.x) >> 5;
    long nw  = ((long)gridDim.x * blockDim.x) >> 5;
    for (long e = wid; e < E; e += nw) {
        const int r = rows[e];
        const int c = cols[e];
        long en = e + nw;
        if (en < E) {
            int rn = rows[en];
            __builtin_prefetch(h + (long)rn * D + lane * 4, 0, 0);
        }
        const float norm = dinv[r] * dinv[c];
        const float4 hv = reinterpret_cast<const float4*>(h + (long)r * D)[lane];
        float* op = out + (long)c * D + lane * 4;
        atomicAdd(op + 0, norm * hv.x);
        atomicAdd(op + 1, norm * hv.y);
        atomicAdd(op + 2, norm * hv.z);
        atomicAdd(op + 3, norm * hv.w);
    }
}

// ---------------------------------------------------------------------------
// Epilogue: acc += dinv[i]^2 * h (self-loop term) + bias; optional ReLU.
// ---------------------------------------------------------------------------
__global__ void epilogue_kernel(float* __restrict__ acc, const float* __restrict__ h,
                                const float* __restrict__ dinv,
                                const float* __restrict__ b, long n, int relu) {
    long idx = (long)blockIdx.x * blockDim.x + threadIdx.x;
    long stride = (long)gridDim.x * blockDim.x;
    for (; idx < n; idx += stride) {
        long i = idx >> 7;
        int  j = (int)(idx & (D - 1));
        float di = dinv[i];
        float v = acc[idx] + di * di * h[idx] + b[j];
        if (relu) v = fmaxf(v, 0.0f);
        acc[idx] = v;
    }
}

// ---------------------------------------------------------------------------
// Launch
// ---------------------------------------------------------------------------
extern "C" void kernel_launch(void* const* d_in, const int* in_sizes, int n_in,
                              void* d_out, int out_size, void* d_ws, size_t ws_size,
                              hipStream_t stream) {
    (void)n_in; (void)out_size; (void)ws_size;

    const float* x_d  = (const float*)d_in[0];
    const float* x_p  = (const float*)d_in[1];
    const int*   e_d  = (const int*)d_in[2];
    const int*   e_p  = (const int*)d_in[3];
    const float* W_d0 = (const float*)d_in[4];
    const float* b_d0 = (const float*)d_in[5];
    const float* W_p0 = (const float*)d_in[6];
    const float* b_p0 = (const float*)d_in[7];
    const float* W_d1 = (const float*)d_in[8];
    const float* b_d1 = (const float*)d_in[9];
    const float* W_p1 = (const float*)d_in[10];
    const float* b_p1 = (const float*)d_in[11];

    const long ND = in_sizes[0] / D;   // 200000
    const long NP = in_sizes[1] / D;   // 100000
    const long ED = in_sizes[2] / 2;   // 3.2M
    const long EP = in_sizes[3] / 2;   // 1.6M

    // Workspace layout (floats): dinv_d | dinv_p | hbuf (ND*D) | t0 (ND*D)
    float* ws     = (float*)d_ws;
    float* dinv_d = ws;
    float* dinv_p = dinv_d + ND;
    float* hbuf   = dinv_p + NP;
    float* t0     = hbuf + ND * D;

    float* out_d = (float*)d_out;
    float* out_p = out_d + ND * D;

    const int TB = 256;
    const int GS = 2048;

    // Degrees (self-loop contributes 1), then dinv = rsqrt(deg)
    fill_kernel<<<GS, TB, 0, stream>>>(dinv_d, ND, 1.0f);
    fill_kernel<<<GS, TB, 0, stream>>>(dinv_p, NP, 1.0f);
    degree_kernel<<<GS, TB, 0, stream>>>(e_d + ED, ED, dinv_d);  // edge[1] = col
    degree_kernel<<<GS, TB, 0, stream>>>(e_p + EP, EP, dinv_p);
    rsqrt_kernel<<<GS, TB, 0, stream>>>(dinv_d, ND);
    rsqrt_kernel<<<GS, TB, 0, stream>>>(dinv_p, NP);

    auto run_type = [&](const float* x, const int* er, const int* ec,
                        const float* dinv, const float* W0, const float* b0,
                        const float* W1, const float* b1, float* out,
                        long N, long E) {
        const int gemmGrid = (int)((N + 127) / 128);
        // Layer 0
        gemm_wmma_f32_kernel<<<gemmGrid, TB, 0, stream>>>(x, W0, hbuf, (int)N);
        fill_kernel<<<GS, TB, 0, stream>>>(t0, N * D, 0.0f);
        scatter_kernel<<<GS, TB, 0, stream>>>(hbuf, er, ec, dinv, t0, E);
        epilogue_kernel<<<GS, TB, 0, stream>>>(t0, hbuf, dinv, b0, N * D, 1);
        // Layer 1
        gemm_wmma_f32_kernel<<<gemmGrid, TB, 0, stream>>>(t0, W1, hbuf, (int)N);
        fill_kernel<<<GS, TB, 0, stream>>>(out, N * D, 0.0f);
        scatter_kernel<<<GS, TB, 0, stream>>>(hbuf, er, ec, dinv, out, E);
        epilogue_kernel<<<GS, TB, 0, stream>>>(out, hbuf, dinv, b1, N * D, 0);
    };

    run_type(x_d, e_d, e_d + ED, dinv_d, W_d0, b_d0, W_d1, b_d1, out_d, ND, ED);
    run_type(x_p, e_p, e_p + EP, dinv_p, W_p0, b_p0, W_p1, b_p1, out_p, NP, EP);
}